// CPUSelectiveScanMixer_53309134078221
// MI455X (gfx1250) — compile-verified
//
#include <hip/hip_runtime.h>
#include <cstdint>
#include <cstddef>

// ---------------- problem dimensions (from reference) ----------------
#define D_MODEL  768
#define D_STATE  8
#define D_CONV   4
#define D_INNER  1536            // EXPAND * D_MODEL
#define DT_RANK  48
#define NB       8
#define SEQ      1024
#define BS       (NB * SEQ)      // 8192 "rows" (flattened batch*seq)
#define XZ_N     (2 * D_INNER)   // 3072
#define DTBC_N   (DT_RANK + 2 * D_STATE)  // 64

typedef float v2f __attribute__((ext_vector_type(2)));
typedef float v8f __attribute__((ext_vector_type(8)));

#define EPI_NONE      0
#define EPI_SOFTPLUS  1   // softplus(v + bias) + 1e-4
#define EPI_TANH_FROM 2   // tanh(v) for columns >= tanh_from

__device__ __forceinline__ float apply_epilogue(float v, int mode, int n, int tanh_from,
                                                const float* __restrict__ bias) {
  if (bias) v += bias[n];
  if (mode == EPI_SOFTPLUS) {
    v = (v > 20.0f) ? v : log1pf(__expf(v));
    v += 1e-4f;
  } else if (mode == EPI_TANH_FROM) {
    if (n >= tanh_from) v = tanhf(v);
  }
  return v;
}

// K-chunk staged in LDS per block (A tile shared by all 8 waves)
#define KC  64               // K chunk length (floats)
#define KP  66               // padded LDS row stride (floats) -> conflict-free frag reads

// ------------------------------------------------------------------
// C[M,N] = A[M,K] (row stride lda) * W[N,K]^T  (+bias, epilogue)
// fp32 WMMA 16x16x4, 2x2 register tiling: each wave owns a 32x32 C tile.
// blockDim = 256 (8 waves); all waves share rows m0..m0+31, so the A tile
// is staged once per block into LDS via async global->LDS copies
// (double-buffered K chunks), and frags are read back with ds_load_b64.
// B (weights) stays on the direct global path: it is L2-resident and
// each wave owns a distinct 32-row slice.
// ------------------------------------------------------------------
__global__ __launch_bounds__(256)
void wmma_gemm_f32_2x2(const float* __restrict__ A, int lda,
                       const float* __restrict__ W,       // (N,K) row-major
                       const float* __restrict__ bias,    // len N or nullptr
                       float* __restrict__ C, int ldc,
                       int M, int N, int K,
                       int mode, int tanh_from) {
  __shared__ float Atile[2][32 * KP];

  const int lane = threadIdx.x & 31;
  const int wave = threadIdx.x >> 5;
  const int m0 = blockIdx.x * 32;
  const int n0 = (blockIdx.y * 8 + wave) * 32;
  const bool active = (n0 < N);              // wave-uniform

  const int half = lane >> 4;                // 0: K lo pair, 1: K hi pair
  const int l    = lane & 15;

  const float* W0 = W + (size_t)(n0 + l) * K + half * 2;
  const float* W1 = W0 + (size_t)16 * K;

  v8f acc00 = {}; v8f acc01 = {}; v8f acc10 = {}; v8f acc11 = {};

  const int nchunks = (K + KC - 1) / KC;

  // async copy of one K-chunk of the 32-row A tile into LDS buffer `buf`
  auto copy_chunk = [&](int kc, int buf) {
    const int len   = (K - kc < KC) ? (K - kc) : KC;   // even (K % 2 == 0)
    const int c8max = len >> 1;                        // b64 units per row
    const int units = 32 * c8max;
    for (int u = 0;; ++u) {
      const int unit = (int)threadIdx.x + u * 256;
      if (unit >= units) break;
      const int row = unit / c8max;
      const int c8  = unit % c8max;
      const float* g = A + (size_t)(m0 + row) * lda + kc + c8 * 2;
      const uint32_t ldsa =
          (uint32_t)(uintptr_t)&Atile[buf][row * KP + c8 * 2];
      asm volatile("global_load_async_to_lds_b64 %0, %1, off"
                   :: "v"(ldsa), "v"(g)
                   : "memory");
    }
  };

  copy_chunk(0, 0);
  asm volatile("s_wait_asynccnt 0x0" ::: "memory");
  __syncthreads();

  for (int c = 0; c < nchunks; ++c) {
    const int kc  = c * KC;
    const int len = (K - kc < KC) ? (K - kc) : KC;
    const int buf = c & 1;

    if (c + 1 < nchunks) copy_chunk(kc + KC, buf ^ 1);   // overlap next copy

    if (active) {
      const float* __restrict__ L0 = &Atile[buf][l * KP + half * 2];
      const float* __restrict__ L1 = &Atile[buf][(l + 16) * KP + half * 2];
      #pragma unroll 4
      for (int kk = 0; kk < len; kk += 4) {
        v2f a0 = *(const v2f*)(L0 + kk);                 // ds_load_b64
        v2f a1 = *(const v2f*)(L1 + kk);
        v2f b0 = *(const v2f*)(W0 + kc + kk);            // global b64 (L2-resident)
        v2f b1 = *(const v2f*)(W1 + kc + kk);
        acc00 = __builtin_amdgcn_wmma_f32_16x16x4_f32(false, a0, false, b0, (short)0, acc00, false, false);
        acc01 = __builtin_amdgcn_wmma_f32_16x16x4_f32(false, a0, false, b1, (short)0, acc01, false, false);
        acc10 = __builtin_amdgcn_wmma_f32_16x16x4_f32(false, a1, false, b0, (short)0, acc10, false, false);
        acc11 = __builtin_amdgcn_wmma_f32_16x16x4_f32(false, a1, false, b1, (short)0, acc11, false, false);
      }
    }

    asm volatile("s_wait_asynccnt 0x0" ::: "memory");    // next chunk landed
    __syncthreads();                                     // all waves done with `buf`
  }

  if (!active) return;

  #pragma unroll
  for (int r = 0; r < 8; ++r) {
    const int mr = r + half * 8;
    int m = m0 + mr;
    int n = n0 + l;
    C[(size_t)m * ldc + n] = apply_epilogue(acc00[r], mode, n, tanh_from, bias);
    n = n0 + 16 + l;
    C[(size_t)m * ldc + n] = apply_epilogue(acc01[r], mode, n, tanh_from, bias);
    m = m0 + 16 + mr;
    n = n0 + l;
    C[(size_t)m * ldc + n] = apply_epilogue(acc10[r], mode, n, tanh_from, bias);
    n = n0 + 16 + l;
    C[(size_t)m * ldc + n] = apply_epilogue(acc11[r], mode, n, tanh_from, bias);
  }
}

// ------------------------------------------------------------------
// Depthwise causal conv (width 4, left pad 3) + bias + SiLU.
// Reads x_part = xz[:, 0:D_INNER]; writes xa (BS, D_INNER).
// ------------------------------------------------------------------
__global__ __launch_bounds__(256)
void conv_silu_kernel(const float* __restrict__ xz,
                      const float* __restrict__ conv_w,   // (D_INNER, 1, D_CONV)
                      const float* __restrict__ conv_b,   // (D_INNER)
                      float* __restrict__ xa) {
  size_t idx = (size_t)blockIdx.x * blockDim.x + threadIdx.x;
  if (idx >= (size_t)BS * D_INNER) return;
  const int    i   = (int)(idx % D_INNER);
  const size_t row = idx / D_INNER;          // b*SEQ + s
  const int    s   = (int)(row % SEQ);

  float acc = conv_b[i];
  #pragma unroll
  for (int k = 0; k < D_CONV; ++k) {
    const int sp = s - (D_CONV - 1) + k;
    if (sp >= 0)
      acc += conv_w[i * D_CONV + k] * xz[(row - (D_CONV - 1) + k) * (size_t)XZ_N + i];
  }
  xa[idx] = acc / (1.0f + __expf(-acc));     // silu
}

// ------------------------------------------------------------------
// Selective scan, lane-parallel over D_STATE: 8 lanes per recurrence,
// 4 recurrences per wave -> 3072 waves (8x the occupancy of 1 thread
// per channel) and the per-step critical path is 1 exp + a few fmas.
// dt lives in xz[:, 0:D_INNER] (written by the dt GEMM), z in xz[:, D_INNER:].
// b/c terms are tanh'd columns 48..63 of dtbc. y written in-place over xa.
// ------------------------------------------------------------------
__global__ __launch_bounds__(256)
void scan_silu_kernel(const float* __restrict__ xa,
                      const float* __restrict__ xz,       // dt | z
                      const float* __restrict__ dtbc,
                      const float* __restrict__ A_log,    // (D_INNER, D_STATE)
                      const float* __restrict__ D_skip,   // (D_INNER)
                      float* __restrict__ y) {
  const int tid  = blockIdx.x * blockDim.x + threadIdx.x;
  const int lane = tid & 31;
  const int q    = lane >> 3;                // quad (channel slot in wave): 0..3
  const int n    = lane & 7;                 // state index owned by this lane
  const int ch   = (tid >> 5) * 4 + q;       // global channel: 0 .. NB*D_INNER-1
  if (ch >= NB * D_INNER) return;
  const int i = ch % D_INNER;
  const int b = ch / D_INNER;

  const float a   = -__expf(A_log[i * D_STATE + n]);
  const float dsk = D_skip[i];
  float st = 0.0f;

  for (int s = 0; s < SEQ; ++s) {
    const size_t row = (size_t)b * SEQ + s;
    const float x_t  = xa[row * D_INNER + i];            // broadcast in quad
    const float dt_t = xz[row * XZ_N + i];               // broadcast in quad
    const float bt   = dtbc[row * DTBC_N + DT_RANK + n];
    const float ct   = dtbc[row * DTBC_N + DT_RANK + D_STATE + n];

    const float da = __expf(dt_t * a);
    st = da * st + (1.0f - da) * bt * x_t;

    // reduce st*ct over the 8 states of this quad
    float p = st * ct;
    p += __shfl_xor(p, 1);
    p += __shfl_xor(p, 2);
    p += __shfl_xor(p, 4);

    if (n == 0) {
      const float z   = xz[row * XZ_N + D_INNER + i];
      const float sil = z / (1.0f + __expf(-z));
      y[row * D_INNER + i] = (p + dsk * x_t) * sil;
    }
  }
}

// ------------------------------------------------------------------
extern "C" void kernel_launch(void* const* d_in, const int* in_sizes, int n_in,
                              void* d_out, int out_size, void* d_ws, size_t ws_size,
                              hipStream_t stream) {
  const float* x      = (const float*)d_in[0];
  const float* W_in   = (const float*)d_in[1];
  const float* conv_w = (const float*)d_in[2];
  const float* conv_b = (const float*)d_in[3];
  const float* W_x    = (const float*)d_in[4];
  const float* W_dt   = (const float*)d_in[5];
  const float* b_dt   = (const float*)d_in[6];
  const float* A_log  = (const float*)d_in[7];
  const float* D_skip = (const float*)d_in[8];
  const float* W_out  = (const float*)d_in[9];
  float* out = (float*)d_out;

  // workspace layout (floats): xz | xa | dtbc  (~153 MB)
  float* ws   = (float*)d_ws;
  float* xz   = ws;
  float* xa   = xz + (size_t)BS * XZ_N;
  float* dtbc = xa + (size_t)BS * D_INNER;

  const dim3 blk(256);

  // 1) xz = x @ W_in^T         (8192 x 3072, K=768)
  wmma_gemm_f32_2x2<<<dim3(BS / 32, (XZ_N / 32 + 7) / 8), blk, 0, stream>>>(
      x, D_MODEL, W_in, nullptr, xz, XZ_N, BS, XZ_N, D_MODEL, EPI_NONE, 0);

  // 2) xa = silu(depthwise_conv(xz[:, :D_INNER]) + conv_b)
  {
    const size_t total = (size_t)BS * D_INNER;
    conv_silu_kernel<<<dim3((unsigned)((total + 255) / 256)), blk, 0, stream>>>(
        xz, conv_w, conv_b, xa);
  }

  // 3) dtbc = xa @ W_x^T       (8192 x 64, K=1536); tanh on cols 48..63
  wmma_gemm_f32_2x2<<<dim3(BS / 32, (DTBC_N / 32 + 7) / 8), blk, 0, stream>>>(
      xa, D_INNER, W_x, nullptr, dtbc, DTBC_N, BS, DTBC_N, D_INNER,
      EPI_TANH_FROM, DT_RANK);

  // 4) dt = softplus(dtbc[:, :48] @ W_dt^T + b_dt) + 1e-4
  //    written into the dead x_part columns of xz (ldc = 3072).
  wmma_gemm_f32_2x2<<<dim3(BS / 32, (D_INNER / 32 + 7) / 8), blk, 0, stream>>>(
      dtbc, DTBC_N, W_dt, b_dt, xz, XZ_N, BS, D_INNER, DT_RANK,
      EPI_SOFTPLUS, 0);

  // 5) selective scan + silu(z) gating; y overwrites xa in place
  //    (8 lanes per recurrence, 4 recurrences per wave)
  scan_silu_kernel<<<dim3((NB * D_INNER * D_STATE) / 256), blk, 0, stream>>>(
      xa, xz, dtbc, A_log, D_skip, xa);

  // 6) out = y @ W_out^T       (8192 x 768, K=1536)
  wmma_gemm_f32_2x2<<<dim3(BS / 32, (D_MODEL / 32 + 7) / 8), blk, 0, stream>>>(
      xa, D_INNER, W_out, nullptr, out, D_MODEL, BS, D_MODEL, D_INNER,
      EPI_NONE, 0);
}